// GNN_Layer_Init_49873160241781
// MI455X (gfx1250) — compile-verified
//
#include <hip/hip_runtime.h>

// GNN layer: out = segment_sum(edge_val * W[edge_col], edge_row) + bias
// N=100000 nodes, D=32 features, E=1.6M edges.
//
// Roofline: 2*E*D ~= 102 MFLOP vs ~45 MB HBM traffic (19 MB edge stream,
// 12.8 MB weight table resident in 192 MB L2, 12.8 MB output) -> ~2 us floor
// at 23.3 TB/s. Pure data-movement problem; WMMA inapplicable (random sparse
// adjacency, no dense tiles). CDNA5 paths used instead:
//   - global_load_async_to_lds_b32 (ASYNCcnt) double-buffered edge staging
//   - s_wait_asynccnt split-counter pipelining
//   - wave32 lane-per-feature (D=32 == one wave): 128B coalesced gathers
//   - native non-returning global_atomic_add_f32 scatter (verified in asm)

#define D_FEAT 32
#define BLOCK  256
#define CHUNK  1024                      // edges staged per block per step
#define NWAVES (BLOCK / 32)
#define NSTAGE ((CHUNK / BLOCK) * 3)     // async instrs per wave per stage (=12)

#if defined(__has_builtin)
#if __has_builtin(__builtin_amdgcn_global_load_async_to_lds_b32)
#define HAS_ASYNC_LDS 1
#endif
#endif
#ifndef HAS_ASYNC_LDS
#define HAS_ASYNC_LDS 0
#endif

// Builtin signature (from clang diagnostic): (global int* src, lds int* dst, imm, imm)
typedef __attribute__((address_space(1))) int as1_int;
typedef __attribute__((address_space(3))) int as3_int;
// Global generic pointers are value-identical to AS1; LDS generic pointers carry
// the byte offset in the low 32 bits (ISA 10.2), so truncation yields the AS3 value.
#define AS1P(p) ((as1_int*)(unsigned long long)(p))
#define AS3P(p) ((as3_int*)(unsigned int)(unsigned long long)(p))

__device__ __forceinline__ void atomic_add_f32(float* addr, float v) {
  // Lowers to: global_atomic_add_f32 vADDR, vDATA, s[base] scale_offset scope:SCOPE_DEV
  __hip_atomic_fetch_add(addr, v, __ATOMIC_RELAXED, __HIP_MEMORY_SCOPE_AGENT);
}

__device__ __forceinline__ void async_wait_le(bool overlapped) {
#if HAS_ASYNC_LDS
#if defined(__has_builtin) && __has_builtin(__builtin_amdgcn_s_wait_asynccnt)
  if (overlapped) __builtin_amdgcn_s_wait_asynccnt(NSTAGE);
  else            __builtin_amdgcn_s_wait_asynccnt(0);
#else
  if (overlapped) asm volatile("s_wait_asynccnt %0" :: "i"(NSTAGE) : "memory");
  else            asm volatile("s_wait_asynccnt 0" ::: "memory");
#endif
#else
  (void)overlapped;
#endif
}

__device__ __forceinline__ void stage_chunk(const int* __restrict__ erow,
                                            const int* __restrict__ ecol,
                                            const float* __restrict__ eval,
                                            int* rS, int* cS, float* vS,
                                            int base, int E) {
#pragma unroll
  for (int k = 0; k < CHUNK / BLOCK; ++k) {
    const int i = (int)threadIdx.x + k * BLOCK;
    int g = base + i;
    if (g > E - 1) g = E - 1;  // clamp: always issue full NSTAGE ops, dup reads in-bounds
#if HAS_ASYNC_LDS
    __builtin_amdgcn_global_load_async_to_lds_b32(AS1P(erow + g), AS3P(rS + i), 0, 0);
    __builtin_amdgcn_global_load_async_to_lds_b32(AS1P(ecol + g), AS3P(cS + i), 0, 0);
    __builtin_amdgcn_global_load_async_to_lds_b32(AS1P(eval + g), AS3P(vS + i), 0, 0);
#else
    rS[i] = erow[g];
    cS[i] = ecol[g];
    vS[i] = eval[g];
#endif
  }
}

__global__ __launch_bounds__(BLOCK) void gnn_scatter_kernel(
    const int* __restrict__ erow, const int* __restrict__ ecol,
    const float* __restrict__ eval, const float* __restrict__ W,
    float* __restrict__ out, int E) {
  __shared__ __align__(16) int   rowS[2][CHUNK];
  __shared__ __align__(16) int   colS[2][CHUNK];
  __shared__ __align__(16) float valS[2][CHUNK];

  const int stride = (int)gridDim.x * CHUNK;
  int base = (int)blockIdx.x * CHUNK;
  if (base >= E) return;  // uniform per block: safe w.r.t. barriers

  const int lane = (int)threadIdx.x & 31;
  const int wave = (int)threadIdx.x >> 5;

  stage_chunk(erow, ecol, eval, rowS[0], colS[0], valS[0], base, E);
  int cur = 0;

  for (; base < E; base += stride, cur ^= 1) {
    const int  nb        = base + stride;
    const bool have_next = nb < E;
    if (have_next)
      stage_chunk(erow, ecol, eval, rowS[cur ^ 1], colS[cur ^ 1], valS[cur ^ 1], nb, E);

    // wait until only the next chunk's NSTAGE ops remain outstanding (in-order)
    async_wait_le(have_next);
    __syncthreads();  // all waves' staging of `cur` visible

    int count = E - base;
    if (count > CHUNK) count = CHUNK;

    for (int i = wave; i < count; i += NWAVES) {
      const int   r = __builtin_amdgcn_readfirstlane(rowS[cur][i]);  // scalar row
      const int   c = __builtin_amdgcn_readfirstlane(colS[cur][i]);  // scalar col
      const float v = valS[cur][i];
      // lane-per-feature: coalesced 128B gather of W row (L2-resident, 12.8MB)
      const float contrib = v * W[c * D_FEAT + lane];
      // native f32 atomic add, device scope, relaxed, non-returning
      atomic_add_f32(&out[r * D_FEAT + lane], contrib);
    }
    __syncthreads();  // done reading `cur` before it is restaged next iteration
  }
}

__global__ __launch_bounds__(BLOCK) void gnn_init_bias_kernel(
    const float* __restrict__ bias, float* __restrict__ out, int n) {
  int i = blockIdx.x * BLOCK + (int)threadIdx.x;
  if (i < n) out[i] = bias[i & (D_FEAT - 1)];
}

extern "C" void kernel_launch(void* const* d_in, const int* in_sizes, int n_in,
                              void* d_out, int out_size, void* d_ws, size_t ws_size,
                              hipStream_t stream) {
  const int*   erow = (const int*)d_in[0];
  const int*   ecol = (const int*)d_in[1];
  const float* eval = (const float*)d_in[2];
  const float* W    = (const float*)d_in[3];
  const float* bias = (const float*)d_in[4];
  float*       out  = (float*)d_out;

  const int E = in_sizes[0];

  // 1) out = broadcast(bias)  (must fully rewrite: harness poisons d_out)
  {
    int grid = (out_size + BLOCK - 1) / BLOCK;
    gnn_init_bias_kernel<<<grid, BLOCK, 0, stream>>>(bias, out, out_size);
  }

  // 2) scatter-accumulate edges
  {
    int chunks = (E + CHUNK - 1) / CHUNK;
    int grid   = chunks < 1024 ? chunks : 1024;  // blocks grid-stride over chunks
    gnn_scatter_kernel<<<grid, BLOCK, 0, stream>>>(erow, ecol, eval, W, out, E);
  }
}